// HeadAttention_79525614453431
// MI455X (gfx1250) — compile-verified
//
#include <hip/hip_runtime.h>
#include <hip/hip_bf16.h>

#define C_EMBD 1024
#define HDIM   64
#define TSEQ   4096
#define NSPLIT 4          // key-dimension split factor for the attention kernel

typedef __attribute__((ext_vector_type(2))) float v2f;
typedef __attribute__((ext_vector_type(4))) float v4f;
typedef __attribute__((ext_vector_type(8))) float v8f;

__device__ __forceinline__ v8f wmma_f32(v2f a, v2f b, v8f c) {
    // D = A(16x4) * B(4x16) + C(16x16), fp32, wave32
    return __builtin_amdgcn_wmma_f32_16x16x4_f32(
        false, a, false, b, (short)0, c, false, false);
}

// ---------------------------------------------------------------------------
// Kernel 0: transpose Wq|Wk|Wv [1024][64] -> Wt[3][64][1024] so that B-matrix
// WMMA fragments become contiguous b64 loads.
// ---------------------------------------------------------------------------
__global__ void transpose_w(const float* __restrict__ Wq,
                            const float* __restrict__ Wk,
                            const float* __restrict__ Wv,
                            float* __restrict__ wt) {
    int i = blockIdx.x * blockDim.x + threadIdx.x;   // 0 .. 3*65536-1
    int m  = i >> 16;
    int e  = i & 65535;
    int n  = e >> 10;      // 0..63 output row (head col)
    int kk = e & 1023;     // 0..1023
    const float* W = (m == 0) ? Wq : ((m == 1) ? Wk : Wv);
    wt[m * 65536 + n * 1024 + kk] = W[kk * 64 + n];
}

// ---------------------------------------------------------------------------
// Kernel 1: fused QKV projection with fp32 WMMA.
// Block = 128 threads = 4 waves; wave w computes rows [blk*64+16w, +16) x 192.
// Q,K stored row-major [16384][64]; V stored transposed Vt[B][64][T].
// ---------------------------------------------------------------------------
__global__ __launch_bounds__(128) void qkv_proj(
    const float* __restrict__ x,  const float* __restrict__ wt,
    const float* __restrict__ bq, const float* __restrict__ bk,
    const float* __restrict__ bv,
    float* __restrict__ qbuf, float* __restrict__ kbuf, float* __restrict__ vt) {

    const int lane = threadIdx.x & 31;
    const int w    = threadIdx.x >> 5;
    const int l15  = lane & 15;
    const int half = lane >> 4;          // 0: k={0,1}, 1: k={2,3}
    const int koff = half * 2;
    const int g0   = blockIdx.x * 64 + w * 16;

    const float* xrow = x + (size_t)(g0 + l15) * C_EMBD;

    v8f c[12];
#pragma unroll
    for (int f = 0; f < 12; ++f) c[f] = (v8f)(0.0f);

    for (int kb = 0; kb < C_EMBD; kb += 4) {
        v2f a = *(const v2f*)(xrow + kb + koff);
#pragma unroll
        for (int f = 0; f < 12; ++f) {
            const float* wcol = wt + (f >> 2) * 65536 + ((f & 3) * 16 + l15) * 1024;
            v2f b = *(const v2f*)(wcol + kb + koff);
            c[f] = wmma_f32(a, b, c[f]);
        }
    }

#pragma unroll
    for (int f = 0; f < 4; ++f) {
        const int col = f * 16 + l15;
        const float biq = bq[col], bik = bk[col], biv = bv[col];
#pragma unroll
        for (int r = 0; r < 8; ++r) {
            const int grow = g0 + r + 8 * half;       // global row (b*T + t)
            qbuf[(size_t)grow * HDIM + col] = c[f]     [r] + biq;
            kbuf[(size_t)grow * HDIM + col] = c[4 + f] [r] + bik;
            const int bidx = grow >> 12;               // batch
            const int t    = grow & (TSEQ - 1);
            vt[((size_t)bidx * HDIM + col) * TSEQ + t] = c[8 + f][r] + biv;
        }
    }
}

// ---------------------------------------------------------------------------
// Kernel 2: causal flash attention with key-dimension split (split-K).
// grid.x = 256 query tiles * NSPLIT chunks; block = 128 threads = 4 waves.
// Each block emits unnormalized O plus per-row (m, l) in base-2 domain.
// Empty chunks naturally store m=-inf, l=0, O=0 (zero-trip loop).
// ---------------------------------------------------------------------------
__global__ __launch_bounds__(128) void flash_attn_partial(
    const float* __restrict__ qbuf, const float* __restrict__ kbuf,
    const float* __restrict__ vt,
    float* __restrict__ po, float* __restrict__ pm, float* __restrict__ pl) {

    __shared__ __align__(16) float Pb[4][16][68];   // wave-private P staging

    const int lane = threadIdx.x & 31;
    const int w    = threadIdx.x >> 5;
    const int l15  = lane & 15;
    const int half = lane >> 4;
    const int koff = half * 2;

    const int qt    = blockIdx.x >> 2;          // query tile 0..255
    const int chunk = blockIdx.x & (NSPLIT - 1);
    const int g0blk = qt * 64;                  // first global row of tile
    const int b     = g0blk >> 12;              // batch
    const int m0    = g0blk & (TSEQ - 1);       // first query row in batch
    const int g0    = g0blk + w * 16;           // first global row of wave
    const int qbase = m0 + w * 16 + 8 * half;   // query row for (r): qbase + r

    const int jdiag  = m0 >> 6;
    const int ntiles = jdiag + 1;
    const int ck     = (ntiles + NSPLIT - 1) >> 2;     // tiles per chunk
    const int jstart = chunk * ck;
    const int jend   = min(jstart + ck, ntiles);

    // Q strip resident as 16 A-fragments
    v2f aq[16];
    const float* qrow = qbuf + (size_t)(g0 + l15) * HDIM;
#pragma unroll
    for (int ks = 0; ks < 16; ++ks) aq[ks] = *(const v2f*)(qrow + ks * 4 + koff);

    const float* kbb = kbuf + (size_t)b * TSEQ * HDIM;
    const float* vtb = vt   + (size_t)b * HDIM * TSEQ;

    v8f o[4];
#pragma unroll
    for (int ns = 0; ns < 4; ++ns) o[ns] = (v8f)(0.0f);
    float m_i[8], l_i[8];
#pragma unroll
    for (int r = 0; r < 8; ++r) { m_i[r] = -__builtin_inff(); l_i[r] = 0.0f; }

    // 1/sqrt(64) * log2(e): softmax computed in base-2 domain (exp2 == v_exp_f32)
    const float SCALE2 = 0.125f * 1.44269504088896340736f;

    for (int j = jstart; j < jend; ++j) {
        // ---- S = Q * K^T  (row-major K is exactly the B-fragment layout) ----
        v8f s[4];
#pragma unroll
        for (int ns = 0; ns < 4; ++ns) {
            v8f acc = (v8f)(0.0f);
            const float* krow = kbb + (size_t)(j * 64 + ns * 16 + l15) * HDIM + koff;
#pragma unroll
            for (int ks = 0; ks < 16; ++ks) {
                v2f bf = *(const v2f*)(krow + ks * 4);
                acc = wmma_f32(aq[ks], bf, acc);
            }
            s[ns] = acc;
        }

        // ---- scale (base-2) + causal mask (diagonal tile only) ----
        const bool diag = (j == jdiag);
#pragma unroll
        for (int ns = 0; ns < 4; ++ns) {
#pragma unroll
            for (int r = 0; r < 8; ++r) {
                float vv = s[ns][r] * SCALE2;
                if (diag) {
                    const int kj = j * 64 + ns * 16 + l15;
                    if (kj > qbase + r) vv = -__builtin_inff();
                }
                s[ns][r] = vv;
            }
        }

        // ---- online softmax (half-wave shfl reductions, wave32) ----
#pragma unroll
        for (int r = 0; r < 8; ++r) {
            float mx = fmaxf(fmaxf(s[0][r], s[1][r]), fmaxf(s[2][r], s[3][r]));
#pragma unroll
            for (int d = 1; d < 16; d <<= 1) mx = fmaxf(mx, __shfl_xor(mx, d, 32));
            const float mnew  = fmaxf(m_i[r], mx);
            const float alpha = exp2f(m_i[r] - mnew);
            m_i[r] = mnew;
            float rs = 0.0f;
#pragma unroll
            for (int ns = 0; ns < 4; ++ns) {
                float p = exp2f(s[ns][r] - mnew);
                s[ns][r] = p;
                rs += p;
                Pb[w][r + 8 * half][ns * 16 + l15] = p;  // C-layout -> LDS
            }
#pragma unroll
            for (int d = 1; d < 16; d <<= 1) rs += __shfl_xor(rs, d, 32);
            l_i[r] = l_i[r] * alpha + rs;
#pragma unroll
            for (int ns = 0; ns < 4; ++ns) o[ns][r] *= alpha;
        }

        // ---- O += P * V  (P re-read from LDS in A-layout; Vt gives B-layout) ----
        v2f ap[16];
#pragma unroll
        for (int ks = 0; ks < 16; ++ks)
            ap[ks] = *(const v2f*)&Pb[w][l15][ks * 4 + koff];
#pragma unroll
        for (int ns = 0; ns < 4; ++ns) {
            const float* vrow = vtb + (size_t)(ns * 16 + l15) * TSEQ + j * 64 + koff;
#pragma unroll
            for (int ks = 0; ks < 16; ++ks) {
                v2f bf = *(const v2f*)(vrow + ks * 4);
                o[ns] = wmma_f32(ap[ks], bf, o[ns]);
            }
        }
    }

    // ---- write unnormalized partials ----
    const int pbase = (qt * NSPLIT + chunk) * 64;    // partial row base
#pragma unroll
    for (int r = 0; r < 8; ++r) {
        const int prow = pbase + w * 16 + r + 8 * half;
#pragma unroll
        for (int ns = 0; ns < 4; ++ns)
            po[(size_t)prow * HDIM + ns * 16 + l15] = o[ns][r];
        if (l15 == 0) {
            pm[prow] = m_i[r];
            pl[prow] = l_i[r];
        }
    }
}

// ---------------------------------------------------------------------------
// Kernel 3: merge NSPLIT partials per query row (log-sum-exp rescale, base 2).
// grid = 256 tiles, 64 threads (one per query row of the tile).
// ---------------------------------------------------------------------------
__global__ __launch_bounds__(64) void attn_combine(
    const float* __restrict__ po, const float* __restrict__ pm,
    const float* __restrict__ pl, float* __restrict__ out) {

    const int qt = blockIdx.x;
    const int r  = threadIdx.x;            // 0..63
    const int grow = qt * 64 + r;

    float m4[NSPLIT], l4[NSPLIT];
#pragma unroll
    for (int c = 0; c < NSPLIT; ++c) {
        m4[c] = pm[(qt * NSPLIT + c) * 64 + r];
        l4[c] = pl[(qt * NSPLIT + c) * 64 + r];
    }
    float mstar = fmaxf(fmaxf(m4[0], m4[1]), fmaxf(m4[2], m4[3]));
    float wc[NSPLIT], lsum = 0.0f;
#pragma unroll
    for (int c = 0; c < NSPLIT; ++c) {
        wc[c] = exp2f(m4[c] - mstar);      // 0 for empty chunks (m=-inf)
        lsum += l4[c] * wc[c];
    }
    const float inv = 1.0f / lsum;

    for (int d = 0; d < HDIM; d += 4) {
        v4f acc = (v4f)(0.0f);
#pragma unroll
        for (int c = 0; c < NSPLIT; ++c) {
            const v4f p = *(const v4f*)&po[((size_t)(qt * NSPLIT + c) * 64 + r) * HDIM + d];
            acc += p * wc[c];
        }
        *(v4f*)&out[(size_t)grow * HDIM + d] = acc * inv;
    }
}

// ---------------------------------------------------------------------------
extern "C" void kernel_launch(void* const* d_in, const int* in_sizes, int n_in,
                              void* d_out, int out_size, void* d_ws, size_t ws_size,
                              hipStream_t stream) {
    (void)in_sizes; (void)n_in; (void)out_size; (void)ws_size;
    const float* x  = (const float*)d_in[0];
    // d_in[1] = causal mask (bool) — structure known analytically, unused
    const float* Wq = (const float*)d_in[2];
    const float* bq = (const float*)d_in[3];
    const float* Wk = (const float*)d_in[4];
    const float* bk = (const float*)d_in[5];
    const float* Wv = (const float*)d_in[6];
    const float* bv = (const float*)d_in[7];

    float* ws = (float*)d_ws;
    float* wt   = ws;                    // 3 * 64 * 1024          = 196608
    float* qbuf = wt   + 3 * 65536;      // 16384 * 64             = 1048576
    float* kbuf = qbuf + 16384 * 64;     // 16384 * 64             = 1048576
    float* vtb  = kbuf + 16384 * 64;     // 4 * 64 * 4096          = 1048576
    float* po   = vtb  + 16384 * 64;     // 256*NSPLIT*64*64       = 4194304
    float* pm   = po   + 256 * NSPLIT * 64 * 64;   // 256*NSPLIT*64 = 65536
    float* pl   = pm   + 256 * NSPLIT * 64;        // 256*NSPLIT*64 = 65536

    transpose_w       <<<768, 256, 0, stream>>>(Wq, Wk, Wv, wt);
    qkv_proj          <<<256, 128, 0, stream>>>(x, wt, bq, bk, bv, qbuf, kbuf, vtb);
    flash_attn_partial<<<256 * NSPLIT, 128, 0, stream>>>(qbuf, kbuf, vtb, po, pm, pl);
    attn_combine      <<<256, 64, 0, stream>>>(po, pm, pl, (float*)d_out);
}